// SpatialTransformerAffine_8675833938224
// MI455X (gfx1250) — compile-verified
//
#include <hip/hip_runtime.h>

// 3D affine spatial transformer (grid_sample, trilinear, border padding).
// Coordinate math on the CDNA5 matrix pipe via V_WMMA_F32_16X16X4_F32:
//   D = A(16x4: scaled affine rows at M=0..2 and M=8..10) x B(4x16: [z,y,x,1]
//   mesh columns for 16 voxels) + C(-0.5 bias rows)  ->  lane i's own
//   (iz,iy,ix) land in its D VGPRs 0..2 (two WMMAs per 32-lane wave).
// Gathers are 4x b64 pair-loads (x0,x0+1 contiguous; border handled by a
// select), unsigned 32-bit indices keep the saddr+voffset addressing mode.
// Write-once output uses non-temporal stores so src stays L2-resident.

typedef __attribute__((ext_vector_type(2))) float v2f;
typedef __attribute__((ext_vector_type(8))) float v8f;
typedef float f2v __attribute__((ext_vector_type(2), aligned(4)));

namespace {
constexpr int kD = 128, kH = 160, kW = 128, kB = 4;
constexpr int kN = kD * kH * kW;          // voxels per batch = 2,621,440 = 5*2^19
constexpr int kBlock = 256;               // 8 waves per block (wave32)
constexpr int kUnroll = 8;
constexpr int kSpan = kBlock * kUnroll;   // 2048 voxels = exactly 16 rows of W
}

__global__ __launch_bounds__(kBlock)
void st_affine_wmma(const float* __restrict__ src,
                    const float* __restrict__ aff,
                    float* __restrict__ out)
{
    const int t      = threadIdx.x;
    const int lane   = t & 31;
    const int waveId = t >> 5;
    const int m      = lane & 15;          // B-matrix column / A-matrix row idx
    const bool hiH   = lane >= 16;

    const unsigned blockStart = (unsigned)blockIdx.x * kSpan;
    const int b = blockStart / kN;                         // constant per block
    const unsigned nb0 = blockStart - (unsigned)b * kN;    // multiple of 2048

    const float* __restrict__ sb = src + (size_t)b * kN;

    // ----- A matrix (16x4 f32): scaled affine rows at M=0..2 and M=8..10 -----
    // lanes 0-15: M=lane, vgpr0=K0, vgpr1=K1; lanes 16-31: M=lane-16,
    // vgpr0=K2, vgpr1=K3.  Row scale folds size/(size-1) normalization.
    const bool valid = (m < 3) || (m >= 8 && m < 11);
    const int  r  = (m < 8) ? m : (m - 8);
    const int  rr = valid ? r : 0;
    const float s = (rr == 1) ? (160.0f / 159.0f) : (128.0f / 127.0f);
    const int  j0 = hiH ? 2 : 0;
    const float ta = aff[b * 12 + rr * 4 + j0];
    const float tb = aff[b * 12 + rr * 4 + j0 + 1];
    v2f A;
    A[0] = valid ? ta * s : 0.0f;
    A[1] = valid ? tb * s : 0.0f;

    // C accumulator: the "-0.5" unnormalize bias on rows 0..2 (and 8..10,
    // which map to the same C/D VGPRs 0..2).
    v8f Cm = {};
    Cm[0] = -0.5f; Cm[1] = -0.5f; Cm[2] = -0.5f;

    // Mesh terms for this lane's B column (voxel base+m / base+16+m).
    // waveId*32+m never crosses a 128 boundary even +16, so x is static and
    // the row offset is it*2 + (waveId>>2).
    const int xo    = (waveId * 32 + m) & (kW - 1);
    const int rdel0 = waveId >> 2;
    const unsigned row0 = nb0 >> 7;            // first row of this block
    const int y0b = (int)(row0 % kH);          // one div/mod per thread, total
    const int z0b = (int)(row0 / kH);

#pragma unroll
    for (int it = 0; it < kUnroll; ++it) {
        // ---- B matrices: columns = mesh [z,y,x,1] of 16 voxels ----
        int yv = y0b + it * 2 + rdel0;
        const int wrap = yv >= kH;
        const int zv = z0b + wrap;
        yv = wrap ? yv - kH : yv;

        const float zf = (float)zv, yf = (float)yv, xf = (float)xo;
        v2f B1, B2;
        B1[0] = hiH ? xf           : zf;   // lanes16-31: K2=x, lanes0-15: K0=z
        B1[1] = hiH ? 1.0f         : yf;   // lanes16-31: K3=1, lanes0-15: K1=y
        B2[0] = hiH ? (xf + 16.0f) : zf;   // second 16 voxels of the wave
        B2[1] = B1[1];

        v8f d1 = __builtin_amdgcn_wmma_f32_16x16x4_f32(
            false, A, false, B1, (short)0, Cm, false, false);
        v8f d2 = __builtin_amdgcn_wmma_f32_16x16x4_f32(
            false, A, false, B2, (short)0, Cm, false, false);

        // Lane L<16: voxel base+L in d1 (M=0..2); L>=16: base+L in d2 (M=8..10).
        float iz = hiH ? d2[0] : d1[0];
        float iy = hiH ? d2[1] : d1[1];
        float ix = hiH ? d2[2] : d1[2];

        // Border padding: clamp before interpolation.
        ix = fminf(fmaxf(ix, 0.0f), (float)(kW - 1));
        iy = fminf(fmaxf(iy, 0.0f), (float)(kH - 1));
        iz = fminf(fmaxf(iz, 0.0f), (float)(kD - 1));

        const float xf0 = floorf(ix), yf0 = floorf(iy), zf0 = floorf(iz);
        const float wx = ix - xf0, wy = iy - yf0, wz = iz - zf0;
        const int x0 = (int)xf0, y0 = (int)yf0, z0 = (int)zf0;

        // Pair-load base: (xe, xe+1) always inside the row; when x0==W-1 the
        // correct clamped pair is (src[W-1], src[W-1]) -> select below.
        const int xe = min(x0, kW - 2);
        const bool xtop = x0 > xe;             // x0 == W-1
        const int y1 = min(y0 + 1, kH - 1);
        const int z1 = min(z0 + 1, kD - 1);

        const unsigned p00 = ((unsigned)(z0 * kH + y0) << 7) + (unsigned)xe;
        const unsigned p01 = ((unsigned)(z0 * kH + y1) << 7) + (unsigned)xe;
        const unsigned p10 = ((unsigned)(z1 * kH + y0) << 7) + (unsigned)xe;
        const unsigned p11 = ((unsigned)(z1 * kH + y1) << 7) + (unsigned)xe;

        const f2v q00 = *(const f2v*)(sb + p00);
        const f2v q01 = *(const f2v*)(sb + p01);
        const f2v q10 = *(const f2v*)(sb + p10);
        const f2v q11 = *(const f2v*)(sb + p11);

        const float c000 = xtop ? q00[1] : q00[0], c001 = q00[1];
        const float c010 = xtop ? q01[1] : q01[0], c011 = q01[1];
        const float c100 = xtop ? q10[1] : q10[0], c101 = q10[1];
        const float c110 = xtop ? q11[1] : q11[0], c111 = q11[1];

        const float c00 = fmaf(c001 - c000, wx, c000);
        const float c01 = fmaf(c011 - c010, wx, c010);
        const float c10 = fmaf(c101 - c100, wx, c100);
        const float c11 = fmaf(c111 - c110, wx, c110);
        const float c0  = fmaf(c01 - c00, wy, c00);
        const float c1  = fmaf(c11 - c10, wy, c10);
        const float res = fmaf(c1 - c0, wz, c0);

        // Write-once output: non-temporal so src stays resident in WGP$/L2.
        __builtin_nontemporal_store(res, out + blockStart + (unsigned)(it * kBlock + t));
    }
}

extern "C" void kernel_launch(void* const* d_in, const int* in_sizes, int n_in,
                              void* d_out, int out_size, void* d_ws, size_t ws_size,
                              hipStream_t stream) {
    const float* src = (const float*)d_in[0];   // [B,1,D,H,W] f32
    const float* aff = (const float*)d_in[1];   // [B,3,4]     f32
    float* out = (float*)d_out;                 // [B,1,D,H,W] f32
    const int total  = kB * kN;                 // 10,485,760
    const int blocks = total / kSpan;           // 5120 (exact)
    st_affine_wmma<<<blocks, kBlock, 0, stream>>>(src, aff, out);
}